// FNO1d_4638564680226
// MI455X (gfx1250) — compile-verified
//
#include <hip/hip_runtime.h>
#include <math.h>

#define S_LEN   8192
#define W_DIM   128
#define B_DIM   32
#define MODES_N 16
#define NUM_G   8
#define NLAYERS 4
#define PI_F    3.14159265358979323846f

typedef __attribute__((ext_vector_type(2))) float v2f;
typedef __attribute__((ext_vector_type(8))) float v8f;

__device__ __forceinline__ v8f wmma_f32_k4(v2f a, v2f b, v8f c) {
  // D(16x16,f32) = A(16x4,f32) * B(4x16,f32) + C ; full fp32 matrix op on CDNA5
  return __builtin_amdgcn_wmma_f32_16x16x4_f32(false, a, false, b, (short)0, c, false, false);
}

__device__ __forceinline__ float gelu_exact(float x) {
  return 0.5f * x * (1.0f + erff(x * 0.70710678118654752f));
}

// ---------------- g_eff: fold Gabor filters + w2 into one scalar per (layer, mode) ---------
// u[4115+m] of linspace(-0.5,0.5,8231) == m/8230 after ifftshift truncation.
__global__ void K_geff(const float* __restrict__ gamma, const float* __restrict__ fwhm,
                       const float* __restrict__ w2, float* __restrict__ geff) {
  int t = threadIdx.x;
  if (t >= NLAYERS * MODES_N) return;
  int l = t >> 4, m = t & 15;
  float u = (float)m / 8230.0f;
  float acc = 0.f;
  for (int k = 0; k < NUM_G; ++k) {
    float fw = fwhm[l * NUM_G + k];
    float t1 = gamma[l * NUM_G + k] * PI_F / (fw + 1e-5f);
    float d = u - fw;
    acc += w2[l * NUM_G + k] * expf(-(t1 * t1) * (d * d));
  }
  geff[l * MODES_N + m] = acc;
}

// ---------------- inverse-DFT basis, transposed [s][32]: cos rows 0..15, sin rows 16..31 ----
__global__ void K_basis(float* __restrict__ Bt) {
  int s = blockIdx.x * blockDim.x + threadIdx.x;
  if (s >= S_LEN) return;
  #pragma unroll
  for (int m = 0; m < MODES_N; ++m) {
    float th = (2.0f * PI_F / (float)S_LEN) * (float)((s * m) & (S_LEN - 1));
    float sv, cv;
    sincosf(th, &sv, &cv);
    Bt[s * 32 + m] = cv;
    Bt[s * 32 + 16 + m] = sv;
  }
}

// ---------------- fc0: [x, grid] @ fc0_w.T + fc0_b  ->  h[b][c][s] ------------------------
__global__ void K_fc0(const float* __restrict__ x, const float* __restrict__ w,
                      const float* __restrict__ bb, float* __restrict__ hout) {
  int idx = blockIdx.x * 256 + threadIdx.x;           // idx = (b<<20)|(c<<13)|s
  int rem = idx & ((1 << 20) - 1);
  int b = idx >> 20;
  int c = rem >> 13;
  int s = rem & (S_LEN - 1);
  float xv = x[(b << 13) + s];
  float g = (float)s * (1.0f / (float)(S_LEN - 1));
  hout[idx] = w[c * 2] * xv + w[c * 2 + 1] * g + bb[c];
}

// ---------------- forward truncated DFT: Xf[b][m][i] = sum_s h[b][i][s] e^{-2pi i m s/S} ----
__device__ __forceinline__ float waveReduce(float v) {
  v += __shfl_down(v, 16, 32);
  v += __shfl_down(v, 8, 32);
  v += __shfl_down(v, 4, 32);
  v += __shfl_down(v, 2, 32);
  v += __shfl_down(v, 1, 32);
  return v;
}

__global__ void __launch_bounds__(256) K_dft(const float* __restrict__ h,
                                             float* __restrict__ Xre, float* __restrict__ Xim) {
  __shared__ float red[8][32];
  const int i = blockIdx.x;
  const int b = blockIdx.y;
  const int t = threadIdx.x;
  const int lane = t & 31, wave = t >> 5;
  const float* hp = h + ((size_t)b * W_DIM + i) * S_LEN;

  float xre[MODES_N], xim[MODES_N];
  #pragma unroll
  for (int m = 0; m < MODES_N; ++m) { xre[m] = 0.f; xim[m] = 0.f; }

  // w = e^{-2pi i s/S} for s = t; advance by e^{-i pi/32} each step
  float th0 = -2.0f * PI_F * (float)t / (float)S_LEN;
  float wr, wi;
  sincosf(th0, &wi, &wr);
  const float er = 0.99518472667219688624f;   // cos(pi/32)
  const float ei = -0.09801714032956060199f;  // -sin(pi/32)

  for (int it = 0; it < S_LEN / 256; ++it) {
    float hv = hp[t + it * 256];
    float cr = 1.f, ci = 0.f;                 // e^{-i m theta}, m=0
    #pragma unroll
    for (int m = 0; m < MODES_N; ++m) {
      xre[m] = fmaf(hv, cr, xre[m]);
      xim[m] = fmaf(hv, ci, xim[m]);
      float nr = cr * wr - ci * wi;
      ci = cr * wi + ci * wr;
      cr = nr;
    }
    float nwr = wr * er - wi * ei;
    wi = wr * ei + wi * er;
    wr = nwr;
  }

  #pragma unroll
  for (int m = 0; m < MODES_N; ++m) {
    float r = waveReduce(xre[m]);
    float q = waveReduce(xim[m]);
    if (lane == 0) { red[wave][m] = r; red[wave][16 + m] = q; }
  }
  __syncthreads();
  if (t < 32) {
    float v = 0.f;
    #pragma unroll
    for (int wv = 0; wv < 8; ++wv) v += red[wv][t];
    int m = t & 15;
    if (t < 16) Xre[((size_t)b * MODES_N + m) * W_DIM + i] = v;
    else        Xim[((size_t)b * MODES_N + m) * W_DIM + i] = v;
  }
}

// ---------------- mode mixing: Z[b][o][0..31] = scaled complex (Xf @ w1) * g_eff ------------
__global__ void __launch_bounds__(256) K_mix(const float* __restrict__ Xre,
                                             const float* __restrict__ Xim,
                                             const float* __restrict__ w1re,
                                             const float* __restrict__ w1im,
                                             const float* __restrict__ geff,
                                             float* __restrict__ Z) {
  const int t = threadIdx.x;
  const int m = t & 15;
  const int ol = t >> 4;                       // 0..15
  const int o = blockIdx.x * 16 + ol;
  const int b = blockIdx.y;

  float re = 0.f, im = 0.f;
  for (int i = 0; i < W_DIM; ++i) {
    float ar = Xre[((size_t)b * MODES_N + m) * W_DIM + i];
    float ai = Xim[((size_t)b * MODES_N + m) * W_DIM + i];
    float wr = w1re[((size_t)i * W_DIM + o) * MODES_N + m];
    float wi = w1im[((size_t)i * W_DIM + o) * MODES_N + m];
    re += ar * wr - ai * wi;
    im += ar * wi + ai * wr;
  }
  float g = geff[m];
  float sc = (m == 0) ? (1.0f / (float)S_LEN) : (2.0f / (float)S_LEN);
  float* zp = Z + ((size_t)b * W_DIM + o) * 32;
  zp[m] = g * sc * re;                                   // cos coefficient
  zp[16 + m] = (m == 0) ? 0.f : (-g * (2.0f / (float)S_LEN) * im);  // sin coefficient
}

// ---------------- fused layer: WMMA conv (K=128) + WMMA inverse-DFT (K=32) + bias + GELU ----
__global__ void __launch_bounds__(256) K_layer(const float* __restrict__ h,
                                               const float* __restrict__ Z,
                                               const float* __restrict__ cw,
                                               const float* __restrict__ cb,
                                               const float* __restrict__ Bt,
                                               float* __restrict__ hout, int applyGelu) {
  __shared__ float lsh[128 * 36];  // h chunk transposed [s=128][i=32], stride 36 (bank-safe)
  __shared__ float lsb[128 * 36];  // basis tile        [s=128][j=32], stride 36
  const int t = threadIdx.x;
  const int b = blockIdx.y;
  const int s0 = blockIdx.x * 128;
  const int lane = t & 31, wave = t >> 5;
  const int ln = lane & 15, hf = lane >> 4;
  const int o0 = wave * 16;

  // stage basis tile (coalesced rows of 32)
  #pragma unroll
  for (int r = 0; r < 16; ++r) {
    int flat = r * 256 + t;
    int ss = flat >> 5, j = flat & 31;
    lsb[ss * 36 + j] = Bt[(size_t)(s0 + ss) * 32 + j];
  }

  v8f acc[8];
  v8f vz = {};
  #pragma unroll
  for (int nt = 0; nt < 8; ++nt) acc[nt] = vz;

  const float* hb = h + (size_t)b * W_DIM * S_LEN;

  // 1x1 conv: K = 128 in 4 staged chunks of 32
  for (int kc = 0; kc < 4; ++kc) {
    __syncthreads();
    #pragma unroll
    for (int r = 0; r < 16; ++r) {
      int flat = r * 256 + t;
      int ii = flat >> 7, ss = flat & 127;   // coalesced along s
      lsh[ss * 36 + ii] = hb[(size_t)(kc * 32 + ii) * S_LEN + s0 + ss];
    }
    __syncthreads();
    for (int kt = 0; kt < 8; ++kt) {
      int k = kt * 4;
      int kg = kc * 32 + k;
      float2 a2 = *reinterpret_cast<const float2*>(cw + (size_t)(o0 + ln) * W_DIM + kg + 2 * hf);
      v2f A = {a2.x, a2.y};
      #pragma unroll
      for (int nt = 0; nt < 8; ++nt) {
        float2 b2 = *reinterpret_cast<const float2*>(&lsh[(nt * 16 + ln) * 36 + k + 2 * hf]);
        v2f Bv = {b2.x, b2.y};
        acc[nt] = wmma_f32_k4(A, Bv, acc[nt]);
      }
    }
  }

  // inverse-DFT: K = 32 over cos/sin basis, A = Z[b]
  const float* Zb = Z + (size_t)b * W_DIM * 32;
  for (int jt = 0; jt < 8; ++jt) {
    int j = jt * 4;
    float2 a2 = *reinterpret_cast<const float2*>(Zb + (size_t)(o0 + ln) * 32 + j + 2 * hf);
    v2f A = {a2.x, a2.y};
    #pragma unroll
    for (int nt = 0; nt < 8; ++nt) {
      float2 b2 = *reinterpret_cast<const float2*>(&lsb[(nt * 16 + ln) * 36 + j + 2 * hf]);
      v2f Bv = {b2.x, b2.y};
      acc[nt] = wmma_f32_k4(A, Bv, acc[nt]);
    }
  }

  // epilogue: + cb, optional exact GELU, write [b][o][s]
  float* ho = hout + (size_t)b * W_DIM * S_LEN;
  #pragma unroll
  for (int nt = 0; nt < 8; ++nt) {
    #pragma unroll
    for (int r = 0; r < 8; ++r) {
      int o = o0 + r + 8 * hf;
      float v = acc[nt][r] + cb[o];
      if (applyGelu) v = gelu_exact(v);
      ho[(size_t)o * S_LEN + s0 + nt * 16 + ln] = v;
    }
  }
}

// ---------------- head: WMMA fc1 (K=128) + GELU + fc2 dot reduction -------------------------
__global__ void __launch_bounds__(256) K_head(const float* __restrict__ h,
                                              const float* __restrict__ fc1w,
                                              const float* __restrict__ fc1b,
                                              const float* __restrict__ fc2w,
                                              const float* __restrict__ fc2b,
                                              float* __restrict__ out) {
  __shared__ float lsh[128 * 36];
  __shared__ float lred[8 * 128];
  const int t = threadIdx.x;
  const int b = blockIdx.y;
  const int s0 = blockIdx.x * 128;
  const int lane = t & 31, wave = t >> 5;
  const int ln = lane & 15, hf = lane >> 4;
  const int o0 = wave * 16;

  v8f acc[8];
  v8f vz = {};
  #pragma unroll
  for (int nt = 0; nt < 8; ++nt) acc[nt] = vz;

  const float* hb = h + (size_t)b * W_DIM * S_LEN;

  for (int kc = 0; kc < 4; ++kc) {
    __syncthreads();
    #pragma unroll
    for (int r = 0; r < 16; ++r) {
      int flat = r * 256 + t;
      int ii = flat >> 7, ss = flat & 127;
      lsh[ss * 36 + ii] = hb[(size_t)(kc * 32 + ii) * S_LEN + s0 + ss];
    }
    __syncthreads();
    for (int kt = 0; kt < 8; ++kt) {
      int k = kt * 4;
      int kg = kc * 32 + k;
      float2 a2 = *reinterpret_cast<const float2*>(fc1w + (size_t)(o0 + ln) * W_DIM + kg + 2 * hf);
      v2f A = {a2.x, a2.y};
      #pragma unroll
      for (int nt = 0; nt < 8; ++nt) {
        float2 b2 = *reinterpret_cast<const float2*>(&lsh[(nt * 16 + ln) * 36 + k + 2 * hf]);
        v2f Bv = {b2.x, b2.y};
        acc[nt] = wmma_f32_k4(A, Bv, acc[nt]);
      }
    }
  }

  // z1 = gelu(fc1 out + b); partial fc2 dot over this wave's 16 o-rows
  float part[8];
  #pragma unroll
  for (int nt = 0; nt < 8; ++nt) part[nt] = 0.f;
  #pragma unroll
  for (int nt = 0; nt < 8; ++nt) {
    #pragma unroll
    for (int r = 0; r < 8; ++r) {
      int o = o0 + r + 8 * hf;
      float v = acc[nt][r] + fc1b[o];
      v = gelu_exact(v);
      part[nt] = fmaf(fc2w[o], v, part[nt]);
    }
  }
  #pragma unroll
  for (int nt = 0; nt < 8; ++nt) part[nt] += __shfl_xor(part[nt], 16, 32);
  if (hf == 0) {
    #pragma unroll
    for (int nt = 0; nt < 8; ++nt) lred[wave * 128 + nt * 16 + ln] = part[nt];
  }
  __syncthreads();
  if (t < 128) {
    float v = fc2b[0];
    #pragma unroll
    for (int wv = 0; wv < 8; ++wv) v += lred[wv * 128 + t];
    out[(size_t)b * S_LEN + s0 + t] = v;
  }
}

extern "C" void kernel_launch(void* const* d_in, const int* in_sizes, int n_in,
                              void* d_out, int out_size, void* d_ws, size_t ws_size,
                              hipStream_t stream) {
  (void)in_sizes; (void)n_in; (void)out_size; (void)ws_size;
  const float* x     = (const float*)d_in[0];
  const float* w1_re = (const float*)d_in[1];
  const float* w1_im = (const float*)d_in[2];
  const float* gamma = (const float*)d_in[3];
  const float* fwhm  = (const float*)d_in[4];
  const float* w2    = (const float*)d_in[5];
  const float* cw    = (const float*)d_in[6];
  const float* cb    = (const float*)d_in[7];
  const float* fc0w  = (const float*)d_in[8];
  const float* fc0b  = (const float*)d_in[9];
  const float* fc1w  = (const float*)d_in[10];
  const float* fc1b  = (const float*)d_in[11];
  const float* fc2w  = (const float*)d_in[12];
  const float* fc2b  = (const float*)d_in[13];
  float* out = (float*)d_out;

  char* w = (char*)d_ws;
  float* hA  = (float*)w;  w += (size_t)B_DIM * W_DIM * S_LEN * 4;   // 128 MB
  float* hB  = (float*)w;  w += (size_t)B_DIM * W_DIM * S_LEN * 4;   // 128 MB
  float* Xre = (float*)w;  w += (size_t)B_DIM * MODES_N * W_DIM * 4; // 256 KB
  float* Xim = (float*)w;  w += (size_t)B_DIM * MODES_N * W_DIM * 4; // 256 KB
  float* Z   = (float*)w;  w += (size_t)B_DIM * W_DIM * 32 * 4;      // 512 KB
  float* Bt  = (float*)w;  w += (size_t)S_LEN * 32 * 4;              // 1 MB
  float* gef = (float*)w;  w += 256;

  K_geff <<<1, 64, 0, stream>>>(gamma, fwhm, w2, gef);
  K_basis<<<S_LEN / 256, 256, 0, stream>>>(Bt);
  K_fc0  <<<(B_DIM * W_DIM * S_LEN) / 256, 256, 0, stream>>>(x, fc0w, fc0b, hA);

  float* hc = hA;
  float* hn = hB;
  for (int l = 0; l < NLAYERS; ++l) {
    K_dft  <<<dim3(W_DIM, B_DIM), 256, 0, stream>>>(hc, Xre, Xim);
    K_mix  <<<dim3(W_DIM / 16, B_DIM), 256, 0, stream>>>(Xre, Xim,
              w1_re + (size_t)l * W_DIM * W_DIM * MODES_N,
              w1_im + (size_t)l * W_DIM * W_DIM * MODES_N,
              gef + l * MODES_N, Z);
    K_layer<<<dim3(S_LEN / 128, B_DIM), 256, 0, stream>>>(hc, Z,
              cw + (size_t)l * W_DIM * W_DIM, cb + (size_t)l * W_DIM,
              Bt, hn, (l < NLAYERS - 1) ? 1 : 0);
    float* tmp = hc; hc = hn; hn = tmp;
  }
  K_head <<<dim3(S_LEN / 128, B_DIM), 256, 0, stream>>>(hc, fc1w, fc1b, fc2w, fc2b, out);
}